// JosephProjector3D_45698452030113
// MI455X (gfx1250) — compile-verified
//
#include <hip/hip_runtime.h>

typedef float v2f __attribute__((ext_vector_type(2)));
typedef float v8f __attribute__((ext_vector_type(8)));

namespace {
constexpr int Wd = 128, Hd = 128, Dd = 128;   // volume (W,H,D), D innermost
constexpr int Ud = 128, Vd = 128, Aa = 120, NS = 128;
}

// One wave == one (angle, u) ray. Lane n=lane&15 owns detector-z column slots
// z = 16*j + n (j=0..7 WMMA accumulators, N=16 each -> 128 z).
// Half h=lane>>4 gathers t-steps {t0+2h, t0+2h+1} of each 4-step chunk, placing
// samples in the B operand of V_WMMA_F32_16X16X4_F32; A = DT*ones makes every
// row of D equal DT * sum_k B[k,n]  ==  the exact Joseph line integral.
__global__ __launch_bounds__(256)
void joseph_fwd_wmma(const float* __restrict__ vol,
                     const float* __restrict__ angles,
                     float* __restrict__ out)
{
    const int lane = threadIdx.x & 31;
    const int wv   = threadIdx.x >> 5;
    const int a    = (int)blockIdx.y;
    const int u    = (int)blockIdx.x * 8 + wv;

    const double Td = 63.5 * 1.4142135623730950488;   // T = 0.5*sqrt(2*127^2)
    const float  T  = (float)Td;
    const float  DT = (float)(2.0 * Td / (double)NS);

    const float ang = angles[a];
    const float cs  = cosf(ang);
    const float sn  = sinf(ang);
    const float up  = (float)u - 63.5f;

    const int n = lane & 15;   // B/D column (z mod 16)
    const int h = lane >> 4;   // K half-group

    v2f amat; amat[0] = DT; amat[1] = DT;   // A = DT (folds the *DT scaling)

    v8f acc[8];
    #pragma unroll
    for (int j = 0; j < 8; ++j) {
        #pragma unroll
        for (int e = 0; e < 8; ++e) acc[j][e] = 0.0f;
    }

    // x(t) = xb + t*cs ; y(t) = yb + t*sn
    const float xb = 63.5f - up * sn;
    const float yb = 63.5f + up * cs;

    for (int t0 = 0; t0 < NS; t0 += 4) {
        // Uniform (per-wave) cull of 4-step chunks fully outside the slice.
        const float tA = -T + ((float)t0 + 0.5f) * DT;
        const float tB = -T + ((float)t0 + 3.5f) * DT;
        const float xe0 = fmaf(tA, cs, xb), xe1 = fmaf(tB, cs, xb);
        const float ye0 = fmaf(tA, sn, yb), ye1 = fmaf(tB, sn, yb);
        const float xmn = fminf(xe0, xe1), xmx = fmaxf(xe0, xe1);
        const float ymn = fminf(ye0, ye1), ymx = fmaxf(ye0, ye1);

        v2f bv[8];
        #pragma unroll
        for (int j = 0; j < 8; ++j) { bv[j][0] = 0.0f; bv[j][1] = 0.0f; }

        if (xmx > -1.0f && xmn < (float)Wd && ymx > -1.0f && ymn < (float)Hd) {
            #pragma unroll
            for (int tt = 0; tt < 2; ++tt) {
                const int   t  = t0 + 2 * h + tt;
                const float tp = -T + ((float)t + 0.5f) * DT;
                const float xi = fmaf(tp, cs, xb);
                const float yi = fmaf(tp, sn, yb);
                const float xf = floorf(xi);
                const float yf = floorf(yi);
                const float wx = xi - xf;
                const float wy = yi - yf;
                const int   ix = (int)xf;
                const int   iy = (int)yf;

                float w[4];
                int   col[4];
                #pragma unroll
                for (int k = 0; k < 4; ++k) {
                    const int   cx = ix + (k & 1);
                    const int   cy = iy + (k >> 1);
                    const float wk = ((k & 1) ? wx : 1.0f - wx) *
                                     ((k >> 1) ? wy : 1.0f - wy);
                    const bool ok = (cx >= 0) && (cx < Wd) && (cy >= 0) && (cy < Hd);
                    const int  xc = cx < 0 ? 0 : (cx > Wd - 1 ? Wd - 1 : cx);
                    const int  yc = cy < 0 ? 0 : (cy > Hd - 1 ? Hd - 1 : cy);
                    w[k]   = ok ? wk : 0.0f;
                    col[k] = (xc * Hd + yc) * Dd;          // z contiguous
                }

                // z-columns: half-wave (16 lanes) reads 64B contiguous per corner
                #pragma unroll
                for (int j = 0; j < 8; ++j) {
                    const int z = 16 * j + n;
                    float sv = w[0] * vol[col[0] + z];
                    sv = fmaf(w[1], vol[col[1] + z], sv);
                    sv = fmaf(w[2], vol[col[2] + z], sv);
                    sv = fmaf(w[3], vol[col[3] + z], sv);
                    bv[j][tt] = sv;
                }
            }
        }

        // EXEC is all-ones here (cull only gated the gather).
        #pragma unroll
        for (int j = 0; j < 8; ++j)
            acc[j] = __builtin_amdgcn_wmma_f32_16x16x4_f32(
                false, amat, false, bv[j], (short)0, acc[j], false, false);
    }

    // D layout: lane n (0..15), element 0 == D[M=0, N=n]; all rows identical.
    if (lane < 16) {
        float* o = out + ((size_t)u * Aa + a) * (size_t)Vd;
        #pragma unroll
        for (int j = 0; j < 8; ++j)
            o[16 * j + n] = acc[j][0];
    }
}

extern "C" void kernel_launch(void* const* d_in, const int* in_sizes, int n_in,
                              void* d_out, int out_size, void* d_ws, size_t ws_size,
                              hipStream_t stream) {
    (void)in_sizes; (void)n_in; (void)out_size; (void)d_ws; (void)ws_size;
    const float* vol    = (const float*)d_in[0];   // (1,1,128,128,128) f32
    const float* angles = (const float*)d_in[1];   // (120,) f32
    float*       out    = (float*)d_out;           // (1,1,128,120,128) f32

    dim3 grid(Ud / 8, Aa);   // 8 waves (u values) per 256-thread block
    joseph_fwd_wmma<<<grid, 256, 0, stream>>>(vol, angles, out);
}